// D_constraint1_35845797052726
// MI455X (gfx1250) — compile-verified
//
#include <hip/hip_runtime.h>
#include <math.h>

typedef float v2f __attribute__((ext_vector_type(2)));
typedef float v4f __attribute__((ext_vector_type(4)));
typedef float v8f __attribute__((ext_vector_type(8)));

// Kernel 1: per-wave 16-row x segCols tile. B = ones => WMMA is an
// order-independent row-sum over K, so any bijection of memory columns onto
// K slots is valid. Each lane loads float4 (b128); low half-wave covers
// cols +0..3 of its 8-col group, high half covers +4..7. No shuffles needed.
__global__ void __launch_bounds__(256)
rowsq_wmma_kernel(const float* __restrict__ d, float* __restrict__ segsum,
                  int K, int N, int segCols) {
    const int lane = threadIdx.x & 31;
    const int w    = blockIdx.x * (blockDim.x >> 5) + (threadIdx.x >> 5);
    const int nseg = K / segCols;
    const int tile = w / nseg;
    const int seg  = w % nseg;
    const int row  = (tile << 4) | (lane & 15);
    const int h    = lane >> 4;                     // half-wave select
    const float* base = d + (size_t)row * (size_t)K + seg * segCols + 4 * h;

    const v2f ones = {1.0f, 1.0f};
    v8f c0 = {}, c1 = {}, c2 = {}, c3 = {};        // 4 independent chains

    for (int kk = 0; kk < segCols; kk += 32) {     // 32 cols per iteration
        v4f a0 = *(const v4f*)(base + kk);
        v4f a1 = *(const v4f*)(base + kk + 8);
        v4f a2 = *(const v4f*)(base + kk + 16);
        v4f a3 = *(const v4f*)(base + kk + 24);
        a0 *= a0; a1 *= a1; a2 *= a2; a3 *= a3;
        c0 = __builtin_amdgcn_wmma_f32_16x16x4_f32(false, __builtin_shufflevector(a0, a0, 0, 1), false, ones, (short)0, c0, false, false);
        c1 = __builtin_amdgcn_wmma_f32_16x16x4_f32(false, __builtin_shufflevector(a0, a0, 2, 3), false, ones, (short)0, c1, false, false);
        c2 = __builtin_amdgcn_wmma_f32_16x16x4_f32(false, __builtin_shufflevector(a1, a1, 0, 1), false, ones, (short)0, c2, false, false);
        c3 = __builtin_amdgcn_wmma_f32_16x16x4_f32(false, __builtin_shufflevector(a1, a1, 2, 3), false, ones, (short)0, c3, false, false);
        c0 = __builtin_amdgcn_wmma_f32_16x16x4_f32(false, __builtin_shufflevector(a2, a2, 0, 1), false, ones, (short)0, c0, false, false);
        c1 = __builtin_amdgcn_wmma_f32_16x16x4_f32(false, __builtin_shufflevector(a2, a2, 2, 3), false, ones, (short)0, c1, false, false);
        c2 = __builtin_amdgcn_wmma_f32_16x16x4_f32(false, __builtin_shufflevector(a3, a3, 0, 1), false, ones, (short)0, c2, false, false);
        c3 = __builtin_amdgcn_wmma_f32_16x16x4_f32(false, __builtin_shufflevector(a3, a3, 2, 3), false, ones, (short)0, c3, false, false);
    }

    v8f c = (c0 + c1) + (c2 + c3);
    // C/D layout: VGPR j = row partial sum s[j + 8*h], identical across the
    // 16 lanes of each half. Lane 0 stores rows 0-7, lane 16 stores rows 8-15.
    float* dst = segsum + (size_t)seg * (size_t)N + (size_t)(tile << 4);
    if (lane == 0)       *(v8f*)dst       = c;
    else if (lane == 16) *(v8f*)(dst + 8) = c;
}

// Kernel 2: one thread per row; sum the nseg segment partials, apply (s-1)^2,
// block-reduce (shuffles + LDS), one partial per block. Deterministic.
__global__ void __launch_bounds__(256)
rowloss_kernel(const float* __restrict__ segsum, float* __restrict__ bpartial,
               int N, int nseg) {
    const int row = blockIdx.x * blockDim.x + threadIdx.x;
    float s = 0.0f;
    for (int g = 0; g < nseg; ++g) s += segsum[(size_t)g * (size_t)N + row];
    float e = s - 1.0f;
    float v = e * e;
#pragma unroll
    for (int off = 16; off > 0; off >>= 1) v += __shfl_xor(v, off, 32);
    __shared__ float red[8];
    if ((threadIdx.x & 31) == 0) red[threadIdx.x >> 5] = v;
    __syncthreads();
    if (threadIdx.x < 8) {
        float t = red[threadIdx.x];
#pragma unroll
        for (int off = 4; off > 0; off >>= 1) t += __shfl_xor(t, off, 32);
        if (threadIdx.x == 0) bpartial[blockIdx.x] = t;
    }
}

// Kernel 3: single-wave finish.
__global__ void __launch_bounds__(32)
finalize_kernel(const float* __restrict__ bpartial, int n, float* __restrict__ out) {
    float t = 0.0f;
    for (int i = (int)threadIdx.x; i < n; i += 32) t += bpartial[i];
#pragma unroll
    for (int off = 16; off > 0; off >>= 1) t += __shfl_xor(t, off, 32);
    if (threadIdx.x == 0) out[0] = 0.001f * sqrtf(t);
}

extern "C" void kernel_launch(void* const* d_in, const int* in_sizes, int n_in,
                              void* d_out, int out_size, void* d_ws, size_t ws_size,
                              hipStream_t stream) {
    const float* d = (const float*)d_in[0];
    const int K = 2048;                    // row length per reference
    const int N = in_sizes[0] / K;         // 16384 rows
    const int tiles = N / 16;              // 1024 WMMA row-tiles

    // K-split factor: as many waves as scratch allows (8192 waves at SPLIT=8).
    int SPLIT = 8;
    while (SPLIT > 1 &&
           ((size_t)N * (size_t)SPLIT + 256) * sizeof(float) > ws_size)
        SPLIT >>= 1;
    const int segCols = K / SPLIT;

    float* segsum   = (float*)d_ws;                      // N*SPLIT floats
    float* bpartial = segsum + (size_t)N * (size_t)SPLIT; // N/256 floats

    const int threads = 256;                              // 8 waves/block
    const int blocks1 = (tiles * SPLIT) / (threads / 32);
    const int blocks2 = N / threads;                      // 64

    rowsq_wmma_kernel<<<blocks1, threads, 0, stream>>>(d, segsum, K, N, segCols);
    rowloss_kernel<<<blocks2, threads, 0, stream>>>(segsum, bpartial, N, SPLIT);
    finalize_kernel<<<1, 32, 0, stream>>>(bpartial, blocks2, (float*)d_out);
}